// Sampler_28389733826940
// MI455X (gfx1250) — compile-verified
//
#include <hip/hip_runtime.h>
#include <stdint.h>
#include <math.h>

#define THREADS 256
#define NBINS   2048
#define CAP     2048
#define TOPN    64
#define NLP     20

// gfx1250 async global->LDS path (guarded: falls back to plain vector loads)
#if defined(__gfx1250__) && __has_builtin(__builtin_amdgcn_global_load_async_to_lds_b128) && __has_builtin(__builtin_amdgcn_s_wait_asynccnt)
#define USE_ASYNC 1
typedef int v4i __attribute__((ext_vector_type(4)));
typedef __attribute__((address_space(1))) v4i g_v4i;   // global AS pointee
typedef __attribute__((address_space(3))) v4i l_v4i;   // LDS AS pointee
#define ASYNC_LOAD16(SRC, LDSA)                                              \
  __builtin_amdgcn_global_load_async_to_lds_b128(                            \
      (g_v4i*)(uintptr_t)(SRC), (l_v4i*)(uintptr_t)(LDSA), 0, 0)
#else
#define USE_ASYNC 0
#endif

// Monotonic order-preserving transform: a > b (float) <=> ordkey(a) > ordkey(b)
__device__ __forceinline__ unsigned ordkey(float f) {
  unsigned b = __float_as_uint(f);
  return (b & 0x80000000u) ? ~b : (b | 0x80000000u);
}

__global__ __launch_bounds__(THREADS)
void sampler_topk_topp(const float* __restrict__ logits,
                       const float* __restrict__ temperature,
                       const int*   __restrict__ top_k,
                       const float* __restrict__ top_p,
                       const float* __restrict__ u,
                       int B, int V,
                       int*   __restrict__ out_sampled,
                       float* __restrict__ out_lp,
                       int*   __restrict__ out_idx) {
  const int row = blockIdx.x;
  const int tid = threadIdx.x;
  if (row >= B) return;
  const float* __restrict__ rowlog = logits + (size_t)row * (size_t)V;

  __shared__ unsigned hist[NBINS];
  __shared__ float    cval[CAP];
  __shared__ int      cidx[CAP];
  __shared__ float    red_v[THREADS];
  __shared__ int      red_p[THREADS];
  __shared__ float    sv[TOPN];      // sorted descending values (scaled logits)
  __shared__ int      si[TOPN];      // their vocab indices
  __shared__ float    ework[TOPN];
  __shared__ unsigned s_cnt;
  __shared__ unsigned s_bstar;
  __shared__ int      s_done;
  __shared__ int      s_nsorted;
#if USE_ASYNC
  __shared__ float    stage[2][THREADS * 4];
#endif

  const float t     = temperature[row];
  const float tsafe = (t < 1e-5f) ? 1.0f : t;
  const float inv_t = 1.0f / tsafe;
  int         k     = top_k[row];
  const float p     = top_p[row];

  for (int i = tid; i < NBINS; i += THREADS) hist[i] = 0u;
  if (tid == 0) { s_cnt = 0u; s_done = 0; s_nsorted = TOPN; }
  __syncthreads();

  const int nFull = V / (THREADS * 4);   // full 1024-element tiles

  // ---------------- Pass 1: 2048-bin histogram of top-11 ordered bits -------
#if USE_ASYNC
  {
    const unsigned lds0 = (unsigned)(uintptr_t)&stage[0][tid * 4];
    const unsigned lds1 = (unsigned)(uintptr_t)&stage[1][tid * 4];
    if (nFull > 0) ASYNC_LOAD16(rowlog + (size_t)tid * 4, lds0);
    for (int itr = 0; itr < nFull; ++itr) {
      const int buf = itr & 1;
      if (itr + 1 < nFull) {
        ASYNC_LOAD16(rowlog + (size_t)(itr + 1) * (THREADS * 4) + (size_t)tid * 4,
                     buf ? lds0 : lds1);
        __builtin_amdgcn_s_wait_asynccnt(1);   // previous tile complete (in-order)
      } else {
        __builtin_amdgcn_s_wait_asynccnt(0);
      }
      const float4 w = *(const float4*)&stage[buf][tid * 4];
      atomicAdd(&hist[ordkey(w.x * inv_t) >> 21], 1u);
      atomicAdd(&hist[ordkey(w.y * inv_t) >> 21], 1u);
      atomicAdd(&hist[ordkey(w.z * inv_t) >> 21], 1u);
      atomicAdd(&hist[ordkey(w.w * inv_t) >> 21], 1u);
    }
  }
#else
  for (int itr = 0; itr < nFull; ++itr) {
    const float4 w = ((const float4*)rowlog)[(size_t)itr * THREADS + tid];
    atomicAdd(&hist[ordkey(w.x * inv_t) >> 21], 1u);
    atomicAdd(&hist[ordkey(w.y * inv_t) >> 21], 1u);
    atomicAdd(&hist[ordkey(w.z * inv_t) >> 21], 1u);
    atomicAdd(&hist[ordkey(w.w * inv_t) >> 21], 1u);
  }
#endif
  for (int i = nFull * THREADS * 4 + tid; i < V; i += THREADS)
    atomicAdd(&hist[ordkey(rowlog[i] * inv_t) >> 21], 1u);
  __syncthreads();

  // Find bin b* such that count(key >= b*) >= TOPN (suffix scan from top)
  if (tid == 0) {
    unsigned acc = 0; int b = NBINS - 1;
    for (; b > 0; --b) { acc += hist[b]; if (acc >= TOPN) break; }
    s_bstar = (unsigned)b;
  }
  __syncthreads();
  const unsigned bstar = s_bstar;

  // ---------------- Pass 2: collect candidates (hits L2: 131MB < 192MB) -----
  const int n4 = V >> 2;
  for (int i4 = tid; i4 < n4; i4 += THREADS) {
    __builtin_prefetch(((const float4*)rowlog) + i4 + THREADS, 0, 0);
    const float4 w = ((const float4*)rowlog)[i4];
    const float vv[4] = {w.x * inv_t, w.y * inv_t, w.z * inv_t, w.w * inv_t};
    #pragma unroll
    for (int j = 0; j < 4; ++j) {
      if ((ordkey(vv[j]) >> 21) >= bstar) {
        unsigned pos = atomicAdd(&s_cnt, 1u);
        if (pos < CAP) { cval[pos] = vv[j]; cidx[pos] = i4 * 4 + j; }
      }
    }
  }
  for (int i = n4 * 4 + tid; i < V; i += THREADS) {
    const float v = rowlog[i] * inv_t;
    if ((ordkey(v) >> 21) >= bstar) {
      unsigned pos = atomicAdd(&s_cnt, 1u);
      if (pos < CAP) { cval[pos] = v; cidx[pos] = i; }
    }
  }
  __syncthreads();
  const int nc = (int)(s_cnt < CAP ? s_cnt : CAP);

  // ---------------- Select top-64 (value desc, vocab idx asc on ties) -------
  for (int r = 0; r < TOPN; ++r) {
    float bv = -INFINITY; int bp = -1;
    for (int pos = tid; pos < nc; pos += THREADS) {
      const float v = cval[pos];
      if (v > bv || (v == bv && bp >= 0 && v != -INFINITY && cidx[pos] < cidx[bp])) {
        bv = v; bp = pos;
      }
    }
    red_v[tid] = bv; red_p[tid] = bp;
    __syncthreads();
    for (int s = THREADS / 2; s > 0; s >>= 1) {
      if (tid < s) {
        const float v2 = red_v[tid + s]; const int p2 = red_p[tid + s];
        bool take = false;
        if (p2 >= 0) {
          if (red_p[tid] < 0) take = true;
          else if (v2 > red_v[tid]) take = true;
          else if (v2 == red_v[tid] && cidx[p2] < cidx[red_p[tid]]) take = true;
        }
        if (take) { red_v[tid] = v2; red_p[tid] = p2; }
      }
      __syncthreads();
    }
    if (tid == 0) {
      if (red_p[0] < 0 || red_v[0] == -INFINITY) { s_nsorted = r; s_done = 1; }
      else {
        sv[r] = red_v[0]; si[r] = cidx[red_p[0]];
        cval[red_p[0]] = -INFINITY;          // remove from candidate pool
      }
    }
    __syncthreads();
    if (s_done) break;
  }

  // ---------------- Scalar tail: top-k/top-p, softmax, sample, logprobs -----
  if (tid == 0) {
    int ns = s_nsorted;
    if (ns < 1) ns = 1;
    if (k > ns) k = ns;
    if (k < 1)  k = 1;
    const float kv = sv[k - 1];              // kth largest scaled logit
    int s = k;                               // survivors: sv[j] >= kv (keep ties)
    while (s < ns && sv[s] >= kv) s++;

    const float m0 = sv[0];
    float Z = 0.0f;
    for (int j = 0; j < s; ++j) { ework[j] = expf(sv[j] - m0); Z += ework[j]; }

    // nucleus: ascending cumsum mask <=> keep j iff suffix-sum S_j > (1-p)*Z; keep max
    const float thr = (1.0f - p) * Z;
    int sprime = 1;
    float run = 0.0f;
    for (int j = s - 1; j >= 1; --j) {
      run += ework[j];
      if (run > thr) { sprime = j + 1; break; }
    }

    float Z2 = 0.0f;
    for (int j = 0; j < sprime; ++j) Z2 += ework[j];
    const float logZ2 = logf(Z2);

    // exponential-race sampling: only kept tokens have nonzero prob
    int bestj = 0; float bestScore = -1.0f;
    for (int j = 0; j < sprime; ++j) {
      const float uu = u[(size_t)row * (size_t)V + (size_t)si[j]];
      const float q  = -log1pf(-uu);
      const float sc = ework[j] / q;
      if (sc > bestScore) { bestScore = sc; bestj = j; }
    }
    const int sj          = (t < 1e-5f) ? 0 : bestj;   // greedy = rank-0
    const int sampled_tok = si[sj];
    const float sampled_lp = (sv[sj] - m0) - logZ2;

    out_sampled[row] = sampled_tok;
    float* lp = out_lp  + (size_t)row * (NLP + 1);
    int*   ix = out_idx + (size_t)row * (NLP + 1);
    const int nfill = (sprime < NLP) ? sprime : NLP;
    for (int j = 0; j < nfill; ++j) { lp[j] = (sv[j] - m0) - logZ2; ix[j] = si[j]; }
    // -inf padding: smallest vocab indices not among kept tokens (jax top_k tiebreak)
    int v0 = 0;
    for (int slot = nfill; slot < NLP; ++slot) {
      bool used = true;
      while (used) {
        used = false;
        for (int j = 0; j < sprime; ++j) if (si[j] == v0) { used = true; break; }
        if (used) v0++;
      }
      lp[slot] = -INFINITY; ix[slot] = v0; v0++;
    }
    lp[NLP] = sampled_lp; ix[NLP] = sampled_tok;
  }
}

extern "C" void kernel_launch(void* const* d_in, const int* in_sizes, int n_in,
                              void* d_out, int out_size, void* d_ws, size_t ws_size,
                              hipStream_t stream) {
  (void)n_in; (void)out_size; (void)d_ws; (void)ws_size;
  const float* logits      = (const float*)d_in[0];
  const float* temperature = (const float*)d_in[1];
  const int*   top_k       = (const int*)  d_in[2];
  const float* top_p       = (const float*)d_in[3];
  const float* u           = (const float*)d_in[4];
  const int B = in_sizes[1];
  const int V = in_sizes[0] / B;

  // d_out layout (flat, return order): sampled[B] i32, topk_lp[B,21] f32, topk_idx[B,21] i32
  int*   out_sampled = (int*)d_out;
  float* out_lp      = (float*)d_out + B;
  int*   out_idx     = (int*)((float*)d_out + (size_t)B + (size_t)B * (NLP + 1));

  hipLaunchKernelGGL(sampler_topk_topp, dim3(B), dim3(THREADS), 0, stream,
                     logits, temperature, top_k, top_p, u, B, V,
                     out_sampled, out_lp, out_idx);
}